// TripletLoss_32057635898176
// MI455X (gfx1250) — compile-verified
//
#include <hip/hip_runtime.h>

#define NN 8192
#define DD 128
#define MARGINF 0.5f
#define BIGF 1e30f

typedef __attribute__((ext_vector_type(2))) float v2f;
typedef __attribute__((ext_vector_type(8))) float v8f;

__device__ __forceinline__ v8f wmma_f32_16x16x4(v2f a, v2f b, v8f c) {
  // 8 args: (neg_a, A, neg_b, B, c_mod, C, reuse_a, reuse_b)
  return __builtin_amdgcn_wmma_f32_16x16x4_f32(false, a, false, b, (short)0, c,
                                               false, false);
}

// ---------------- Kernel 1: row squared norms -------------------------------
__global__ void __launch_bounds__(256) k_rownorm(const float* __restrict__ x,
                                                 float* __restrict__ xn) {
  int i = blockIdx.x * blockDim.x + threadIdx.x;
  if (i >= NN) return;
  const float4* row = (const float4*)(x + (size_t)i * DD);
  float s = 0.f;
#pragma unroll
  for (int c = 0; c < DD / 4; ++c) {
    float4 v = row[c];
    s += v.x * v.x + v.y * v.y + v.z * v.z + v.w * v.w;
  }
  xn[i] = s;
}

// ---------------- Kernel 2: fused Gram + triplet mining (WMMA) --------------
// One wave per 16 anchors. Anchors live in WMMA N-dim; candidates stream
// through M-dim in 16-row tiles. Each lane mines its anchor over its half of
// the candidate rows; lane^16 holds the other half; one shfl_xor combines.
__global__ void __launch_bounds__(256) k_mine(const float* __restrict__ x,
                                              const int* __restrict__ tgt,
                                              const float* __restrict__ xn,
                                              int* __restrict__ gpos,
                                              int* __restrict__ gneg,
                                              int* __restrict__ csame,
                                              int* __restrict__ cdiff) {
  const int lane = threadIdx.x & 31;
  const int wave = (int)((blockIdx.x * blockDim.x + threadIdx.x) >> 5);
  const int half = lane >> 4;   // which half of the wave
  const int l16 = lane & 15;
  const int koff = half * 2;    // f32 A/B fragment: K = {0,1} or {2,3} per half

  const int arow = wave * 16 + l16;  // anchor (shared by lanes l16 and l16+16)
  const float xna = xn[arow];
  const int ta = tgt[arow];

  // Register-resident B fragments = anchor row, half per lane across K-chunks.
  v2f bfrag[DD / 4];
#pragma unroll
  for (int c = 0; c < DD / 4; ++c)
    bfrag[c] = *(const v2f*)(x + (size_t)arow * DD + 4 * c + koff);

  float bpv = -BIGF;
  int bpi = half * 8;  // first candidate index of this lane's subset
  float bnv = BIGF;
  int bni = half * 8;
  int cs = 0, cd = 0;

  for (int t = 0; t < NN / 16; ++t) {
    const int crow = t * 16 + l16;  // candidate row for A fragment
    // Prefetch next candidate tile's row into caches (global_prefetch_b8).
    if (t + 1 < NN / 16)
      __builtin_prefetch(x + (size_t)(crow + 16) * DD, 0, 3);

    v8f acc = {};  // C = 0
#pragma unroll
    for (int c = 0; c < DD / 4; ++c) {
      v2f a = *(const v2f*)(x + (size_t)crow * DD + 4 * c + koff);
      acc = wmma_f32_16x16x4(a, bfrag[c], acc);  // G[m=cand][n=anchor]
    }

    // Per-lane epilogue: 8 candidate rows (this half of the tile).
    const int cbase = t * 16 + half * 8;
    const float4 xc0 = *(const float4*)(xn + cbase);
    const float4 xc1 = *(const float4*)(xn + cbase + 4);
    const int4 tc0 = *(const int4*)(tgt + cbase);
    const int4 tc1 = *(const int4*)(tgt + cbase + 4);
    const float xcv[8] = {xc0.x, xc0.y, xc0.z, xc0.w,
                          xc1.x, xc1.y, xc1.z, xc1.w};
    const int tcv[8] = {tc0.x, tc0.y, tc0.z, tc0.w, tc1.x, tc1.y, tc1.z, tc1.w};
#pragma unroll
    for (int r = 0; r < 8; ++r) {
      float dd = xna + xcv[r] - 2.0f * acc[r];
      dd = fmaxf(dd, 0.0f);  // clip >= 0 (diag shift ~ 0)
      const int cidx = cbase + r;
      const bool same = (tcv[r] == ta);
      cs += same ? 1 : 0;
      cd += same ? 0 : 1;
      const bool up_p = same && (dd > bpv);  // strict > => argmax-first
      bpv = up_p ? dd : bpv;
      bpi = up_p ? cidx : bpi;
      const bool up_n = (!same) && (dd < bnv);  // strict < => argmin-first
      bnv = up_n ? dd : bnv;
      bni = up_n ? cidx : bni;
    }
  }

  // Merge the two candidate-index subsets (lane <-> lane^16).
  const float obpv = __shfl_xor(bpv, 16, 32);
  const int obpi = __shfl_xor(bpi, 16, 32);
  const float obnv = __shfl_xor(bnv, 16, 32);
  const int obni = __shfl_xor(bni, 16, 32);
  const int ocs = __shfl_xor(cs, 16, 32);
  const int ocd = __shfl_xor(cd, 16, 32);
  const bool take_p = (obpv > bpv) || (obpv == bpv && obpi < bpi);
  const bool take_n = (obnv < bnv) || (obnv == bnv && obni < bni);
  const int g_pos = take_p ? obpi : bpi;
  const int g_neg = take_n ? obni : bni;

  if (half == 0) {
    gpos[arow] = g_pos;
    gneg[arow] = g_neg;
    csame[arow] = cs + ocs;
    cdiff[arow] = cd + ocd;
  }
}

// ---------------- Kernel 3: subset-rank (faithful index bug) ----------------
__global__ void __launch_bounds__(256) k_rank(const int* __restrict__ tgt,
                                              const int* __restrict__ gpos,
                                              const int* __restrict__ gneg,
                                              int* __restrict__ ppos,
                                              int* __restrict__ pneg) {
  __shared__ int tl[NN];
  for (int j = threadIdx.x; j < NN; j += blockDim.x) tl[j] = tgt[j];
  __syncthreads();
  const int i = blockIdx.x * blockDim.x + threadIdx.x;
  if (i >= NN) return;
  const int ti = tl[i];
  const int gp = gpos[i], gn = gneg[i];
  int cp = 0, cn = 0;
  const int4* t4 = (const int4*)tl;
  for (int j4 = 0; j4 < NN / 4; ++j4) {
    const int4 tv = t4[j4];
    const int j = j4 * 4;
    cp += ((tv.x == ti) && (j + 0 <= gp)) ? 1 : 0;
    cp += ((tv.y == ti) && (j + 1 <= gp)) ? 1 : 0;
    cp += ((tv.z == ti) && (j + 2 <= gp)) ? 1 : 0;
    cp += ((tv.w == ti) && (j + 3 <= gp)) ? 1 : 0;
    cn += ((tv.x != ti) && (j + 0 <= gn)) ? 1 : 0;
    cn += ((tv.y != ti) && (j + 1 <= gn)) ? 1 : 0;
    cn += ((tv.z != ti) && (j + 2 <= gn)) ? 1 : 0;
    cn += ((tv.w != ti) && (j + 3 <= gn)) ? 1 : 0;
  }
  int pp = cp - 1;
  int pn = cn - 1;
  pp = pp < 0 ? 0 : (pp >= NN ? NN - 1 : pp);  // clamp (invalid rows: vf==0)
  pn = pn < 0 ? 0 : (pn >= NN ? NN - 1 : pn);
  ppos[i] = pp;
  pneg[i] = pn;
}

// ---------------- Kernel 4: zero accumulators -------------------------------
__global__ void k_init(float* __restrict__ acc) {
  if (threadIdx.x < 8) acc[threadIdx.x] = 0.0f;
}

// ---------------- Kernel 5: per-row terms + global reduce -------------------
__global__ void __launch_bounds__(256) k_terms(const float* __restrict__ x,
                                               const int* __restrict__ ppos,
                                               const int* __restrict__ pneg,
                                               const int* __restrict__ csame,
                                               const int* __restrict__ cdiff,
                                               float* __restrict__ acc) {
  const int lane = threadIdx.x & 31;
  const int i = (int)((blockIdx.x * blockDim.x + threadIdx.x) >> 5);  // row/wave
  if (i >= NN) return;
  const float4 a = *(const float4*)(x + (size_t)i * DD + lane * 4);
  const float4 p = *(const float4*)(x + (size_t)ppos[i] * DD + lane * 4);
  const float4 q = *(const float4*)(x + (size_t)pneg[i] * DD + lane * 4);

  float ap = 0.f, an = 0.f, nd = 0.f, sa = 0.f, sp = 0.f, sn = 0.f;
  {
    float d;
    d = a.x - p.x; ap += d * d; d = a.y - p.y; ap += d * d;
    d = a.z - p.z; ap += d * d; d = a.w - p.w; ap += d * d;
    d = a.x - q.x; an += d * d; d = a.y - q.y; an += d * d;
    d = a.z - q.z; an += d * d; d = a.w - q.w; an += d * d;
    d = p.x - q.x; nd += d * d; d = p.y - q.y; nd += d * d;
    d = p.z - q.z; nd += d * d; d = p.w - q.w; nd += d * d;
    sa += fabsf(a.x) + fabsf(a.y) + fabsf(a.z) + fabsf(a.w);
    sp += fabsf(p.x) + fabsf(p.y) + fabsf(p.z) + fabsf(p.w);
    sn += fabsf(q.x) + fabsf(q.y) + fabsf(q.z) + fabsf(q.w);
  }
#pragma unroll
  for (int off = 16; off > 0; off >>= 1) {
    ap += __shfl_xor(ap, off, 32);
    an += __shfl_xor(an, off, 32);
    nd += __shfl_xor(nd, off, 32);
    sa += __shfl_xor(sa, off, 32);
    sp += __shfl_xor(sp, off, 32);
    sn += __shfl_xor(sn, off, 32);
  }
  if (lane == 0) {
    const float vf = ((csame[i] > 1) && (cdiff[i] >= 1)) ? 1.0f : 0.0f;
    const float tl = fmaxf(ap - an + MARGINF, 0.0f) * vf;
    atomicAdd(&acc[0], tl);                          // sum(tl)
    atomicAdd(&acc[1], (tl > 0.0f) ? 1.0f : 0.0f);   // non_zero count
    atomicAdd(&acc[2], vf);                          // vcount
    atomicAdd(&acc[3], (sa + sp + sn) * vf);         // L1 sums
    atomicAdd(&acc[4], (ap - an - nd) * vf);         // pairwise numerator
    atomicAdd(&acc[5], ap * vf);                     // sum ap
    atomicAdd(&acc[6], an * vf);                     // sum an
  }
}

// ---------------- Kernel 6: finalize 6 outputs ------------------------------
__global__ void k_final(const float* __restrict__ acc, float* __restrict__ out) {
  if (threadIdx.x != 0 || blockIdx.x != 0) return;
  const float sum_tl = acc[0], nz = acc[1], vcount = acc[2];
  const float sL1 = acc[3], pair = acc[4], sap = acc[5], san = acc[6];
  out[0] = (nz == 0.0f) ? (sum_tl / vcount) : (sum_tl / fmaxf(nz, 1.0f));
  out[1] = sL1 / (3.0f * vcount);
  out[2] = fmaxf(pair / vcount, 0.0f);
  out[3] = vcount;
  out[4] = sap / vcount;
  out[5] = san / vcount;
}

// ---------------- Host-side orchestration -----------------------------------
extern "C" void kernel_launch(void* const* d_in, const int* in_sizes, int n_in,
                              void* d_out, int out_size, void* d_ws,
                              size_t ws_size, hipStream_t stream) {
  const float* x = (const float*)d_in[0];
  const int* tgt = (const int*)d_in[1];
  float* out = (float*)d_out;

  char* ws = (char*)d_ws;
  float* xn = (float*)ws;                               // N floats
  int* gpos = (int*)(ws + sizeof(float) * NN * 1);      // N ints
  int* gneg = (int*)(ws + sizeof(float) * NN * 2);
  int* csame = (int*)(ws + sizeof(float) * NN * 3);
  int* cdiff = (int*)(ws + sizeof(float) * NN * 4);
  int* ppos = (int*)(ws + sizeof(float) * NN * 5);
  int* pneg = (int*)(ws + sizeof(float) * NN * 6);
  float* acc = (float*)(ws + sizeof(float) * NN * 7);   // 8 floats

  k_rownorm<<<NN / 256, 256, 0, stream>>>(x, xn);
  // 512 waves (16 anchors each) = 64 blocks of 256 threads.
  k_mine<<<(NN / 16) * 32 / 256, 256, 0, stream>>>(x, tgt, xn, gpos, gneg,
                                                   csame, cdiff);
  k_rank<<<NN / 256, 256, 0, stream>>>(tgt, gpos, gneg, ppos, pneg);
  k_init<<<1, 32, 0, stream>>>(acc);
  // One wave per row: 8192 waves = 1024 blocks of 256 threads.
  k_terms<<<NN * 32 / 256, 256, 0, stream>>>(x, ppos, pneg, csame, cdiff, acc);
  k_final<<<1, 1, 0, stream>>>(acc, out);
}